// QuanvLayer_89206470738346
// MI455X (gfx1250) — compile-verified
//
#include <hip/hip_runtime.h>
#include <math.h>

#define PI_F 3.14159265358979323846f

// One-thread prep: cos/sin of the 4 circuit weights into d_ws (uniform for all
// patches). Weights are in [0, 2*pi) so the hardware v_cos/v_sin path
// (__cosf/__sinf) is accurate here and avoids the huge scalarized libm
// Payne-Hanek slow path.
__global__ void quanv_trig_prep(const float* __restrict__ w, float* __restrict__ t) {
  if (threadIdx.x == 0 && blockIdx.x == 0) {
    const float w0 = w[0], w1 = w[1], w2 = w[2], w3 = w[3];
    t[0] = __cosf(w0);
    t[1] = __cosf(w1);
    t[2] = __cosf(w2);
    t[3] = __sinf(w2);
    t[4] = __cosf(w3);
    t[5] = __sinf(w3);
  }
}

// One block = 4 pixel rows x 256 cols (4 KB tile) = 2 patch-rows x 128 patch-cols
// = 256 patches, one patch per thread. 4096 blocks cover the whole 16 MB input.
__global__ __launch_bounds__(256) void quanv_main(const float* __restrict__ x,
                                                  const float* __restrict__ trig,
                                                  float* __restrict__ out) {
  __shared__ float tile[1024];  // [4][256] floats, row-major

  const unsigned t   = threadIdx.x;
  const unsigned blk = blockIdx.x;

  // --- CDNA5 async global -> LDS staging (ASYNCcnt path) ---
  // Each lane moves one 16-byte chunk; 256 lanes cover the 4 KB tile.
  const unsigned lds_off = (unsigned)(size_t)(&tile[0]) + t * 16u;
  const unsigned g_off   = blk * 4096u + t * 16u;  // tiles are contiguous in x
  asm volatile("global_load_async_to_lds_b128 %0, %1, %2 offset:0"
               :
               : "v"(lds_off), "v"(g_off), "s"((unsigned long long)(size_t)x)
               : "memory");

  // Uniform weight trig (scalar loads) — overlaps with the async transfer.
  const float cw0 = trig[0];
  const float cw1 = trig[1];
  const float cw2 = trig[2];
  const float sw2 = trig[3];
  const float cw3 = trig[4];
  const float sw3 = trig[5];

  asm volatile("s_wait_asynccnt 0" ::: "memory");
  __syncthreads();

  // --- per-patch closed form (Heisenberg-reduced circuit) ---
  const unsigned pr = t >> 7;    // patch row within tile (0..1)
  const unsigned pc = t & 127u;  // patch col (0..127)
  const float* r0 = &tile[(pr * 2u) * 256u + pc * 2u];
  const float p0 = r0[0], p1 = r0[1];      // top-left, top-right
  const float p2 = r0[256], p3 = r0[257];  // bottom-left, bottom-right

  const float mn = fminf(fminf(p0, p1), fminf(p2, p3));
  const float mx = fmaxf(fmaxf(p0, p1), fmaxf(p2, p3));
  const float rs = PI_F * __builtin_amdgcn_rcpf(mx - mn + 1e-8f);

  const float a0 = (p0 - mn) * rs;
  const float a1 = (p1 - mn) * rs;
  const float a2 = (p2 - mn) * rs;
  const float a3 = (p3 - mn) * rs;

  float s0, c0, s2, c2, s3, c3;
  __sincosf(a0, &s0, &c0);
  const float c1 = __cosf(a1);
  __sincosf(a2, &s2, &c2);
  __sincosf(a3, &s3, &c3);

  const float z0 = cw0 * cw2 * c0 * c2 - sw2 * s0;
  const float z1 = cw1 * c1;
  const float z2 = c2 * c3;
  const float z3 = cw3 * c3 - sw3 * s3 * s2 * s0;

  // out[b, k, i, j]; channel-plane stride 128*128 = 16384
  const unsigned img  = blk >> 6;
  const unsigned prow = ((blk & 63u) << 1) + pr;
  float* o = out + (size_t)img * 65536u + (size_t)prow * 128u + pc;
  o[0]     = z0;
  o[16384] = z1;
  o[32768] = z2;
  o[49152] = z3;
}

extern "C" void kernel_launch(void* const* d_in, const int* in_sizes, int n_in,
                              void* d_out, int out_size, void* d_ws, size_t ws_size,
                              hipStream_t stream) {
  const float* x = (const float*)d_in[0];
  const float* w = (const float*)d_in[1];
  float* out     = (float*)d_out;
  float* trig    = (float*)d_ws;  // 6 floats of scratch

  quanv_trig_prep<<<1, 32, 0, stream>>>(w, trig);
  quanv_main<<<4096, 256, 0, stream>>>(x, trig, out);
}